// ToyModel_56745107915071
// MI455X (gfx1250) — compile-verified
//
#include <hip/hip_runtime.h>
#include <hip/hip_bf16.h>
#include <stdint.h>

// Problem dims
#define INPUT 100
#define HID   275
#define OUTD  100
#define BATCH 256
#define TT    512

// Padded tiling (WMMA 16x16, K-step 4 for f32)
#define NTILES_G 18              // 288 padded cols per gate
#define KK_H 69                  // 276 padded K over hidden
#define KK_X 25                  // 100 K over input (exact)
#define KK_G (KK_H + KK_X)       // 94 fused K iters for gate GEMM
#define NTILES_O 7               // 112 padded output cols
#define WAVES 18
#define NTHREADS (WAVES * 32)

// Workspace layout (floats). Packed so each lane's B-fragment is one b64 load.
#define WG_SZ (4 * NTILES_G * KK_G * 64)   // 433152
#define WO_SZ (NTILES_O * KK_H * 64)       // 30912
#define WG_OFF 0
#define WO_OFF (WG_SZ)
#define BG_OFF (WG_SZ + WO_SZ)             // 464064
#define BG_SZ  (4 * 288)
#define BO_OFF (BG_OFF + BG_SZ)            // 465216
#define BO_SZ  112
#define WS_TOTAL (BO_OFF + BO_SZ)          // 465328 floats (~1.82 MB, L2-resident)

typedef float v2f __attribute__((ext_vector_type(2)));
typedef float v8f __attribute__((ext_vector_type(8)));

// ---- CDNA5 fp32 WMMA: D(16x16,f32) = A(16x4,f32) * B(4x16,f32) + C ----
__device__ __forceinline__ v8f wmma4(v2f a, v2f b, v8f c) {
#if __has_builtin(__builtin_amdgcn_wmma_f32_16x16x4_f32)
  return __builtin_amdgcn_wmma_f32_16x16x4_f32(false, a, false, b, (short)0, c,
                                               false, false);
#else
  c[0] += a[0] * b[0] + a[1] * b[1];
  return c;
#endif
}

// ---- CDNA5 async global->LDS copy (ASYNCcnt-tracked), with fallback ----
#if __has_builtin(__builtin_amdgcn_global_load_async_to_lds_b128)
#define HAVE_ASYNC_LDS 1
typedef int v4i_gcc __attribute__((vector_size(16)));          // matches builtin pointee
typedef __attribute__((address_space(1))) v4i_gcc as1_v4i;     // global (__device__)
typedef __attribute__((address_space(3))) v4i_gcc as3_v4i;     // LDS (__shared__)
__device__ __forceinline__ void async_copy_b128(const float* g, float* l) {
  __builtin_amdgcn_global_load_async_to_lds_b128(
      (as1_v4i*)(uintptr_t)g, (as3_v4i*)(uintptr_t)l, 0, 0);
}
__device__ __forceinline__ void wait_async0() {
#if __has_builtin(__builtin_amdgcn_s_wait_asynccnt)
  __builtin_amdgcn_s_wait_asynccnt(0);
#else
  asm volatile("s_wait_asynccnt 0x0" ::: "memory");
#endif
}
#endif

__device__ __forceinline__ float fast_rcp(float x) {
#if __has_builtin(__builtin_amdgcn_rcpf)
  return __builtin_amdgcn_rcpf(x);
#else
  return 1.0f / x;
#endif
}
__device__ __forceinline__ float sigmoid_f(float x) {
  float e = __expf(-x);
  return fast_rcp(1.0f + e);
}
__device__ __forceinline__ float tanh_f(float x) {
  x = fminf(fmaxf(x, -15.0f), 15.0f);
  float e = __expf(2.0f * x);
  return (e - 1.0f) * fast_rcp(e + 1.0f);
}

// -------------------------------------------------------------------------
// Pack weights/biases into WMMA-fragment order.
// B-fragment layout per (tile j, k-iter kk): 64 floats, lane L holds
//   elem[2L+0] = B[k0 + 2*(L/16) + 0][n = 16*tile + (L%16)]
//   elem[2L+1] = B[k0 + 2*(L/16) + 1][n]
// where B[k][n] = W[row(n)][k]. Gate K is fused: kk<69 -> W_hh, else W_ih.
// -------------------------------------------------------------------------
__global__ void pack_kernel(const float* __restrict__ W_ih,
                            const float* __restrict__ W_hh,
                            const float* __restrict__ b_ih,
                            const float* __restrict__ b_hh,
                            const float* __restrict__ W_out,
                            const float* __restrict__ b_out,
                            float* __restrict__ ws) {
  int idx = blockIdx.x * 256 + threadIdx.x;
  if (idx < WG_SZ) {
    int e  = idx & 63;
    int kk = (idx >> 6) % KK_G;
    int j  = idx / (KK_G * 64);
    int g  = j / NTILES_G, nt = j % NTILES_G;
    int L = e >> 1, half = e & 1;
    int n = nt * 16 + (L & 15);
    int dk = 2 * (L >> 4) + half;
    float v = 0.0f;
    if (n < HID) {
      int r = g * HID + n;               // row in [4H] (gate order i,f,g,o)
      if (kk < KK_H) {
        int k = 4 * kk + dk;
        if (k < HID) v = W_hh[r * HID + k];
      } else {
        int k = 4 * (kk - KK_H) + dk;    // 0..99, always valid
        v = W_ih[r * INPUT + k];
      }
    }
    ws[WG_OFF + idx] = v;
  } else if (idx < WG_SZ + WO_SZ) {
    int i2 = idx - WG_SZ;
    int e  = i2 & 63;
    int kk = (i2 >> 6) % KK_H;
    int j  = i2 / (KK_H * 64);
    int L = e >> 1, half = e & 1;
    int no = j * 16 + (L & 15);
    int k  = 4 * kk + 2 * (L >> 4) + half;
    float v = 0.0f;
    if (no < OUTD && k < HID) v = W_out[no * HID + k];
    ws[idx] = v;
  } else if (idx < BG_OFF + BG_SZ) {
    int n = idx - BG_OFF;
    int g = n / 288, c = n % 288;
    float v = 0.0f;
    if (c < HID) { int r = g * HID + c; v = b_ih[r] + b_hh[r]; }
    ws[idx] = v;
  } else if (idx < WS_TOTAL) {
    int c = idx - BO_OFF;
    ws[idx] = (c < OUTD) ? b_out[c] : 0.0f;
  }
}

// -------------------------------------------------------------------------
// Persistent per-batch-block LSTM. One WG owns 16 batch rows for all 512
// steps. h lives in LDS (zero-padded to K=276), c in accumulator VGPRs.
// Wave w (0..17) owns gate N-tiles {i,f,g,o} x columns [16w,16w+16).
// Waves 0..6 additionally carry one output-projection tile in the same
// K-loop (overlapped), producing out[t-1] while gates for t are computed.
// -------------------------------------------------------------------------
__global__ __launch_bounds__(NTHREADS, 1)
void lstm_kernel(const float* __restrict__ x,
                 const float* __restrict__ ws,
                 float* __restrict__ out) {
  constexpr int HS = 290;                 // LDS row stride (floats), conflict pad
  __shared__ __align__(16) float h_lds[16 * HS];
  __shared__ __align__(16) float x_lds[2][16 * INPUT];

  const int tid   = threadIdx.x;
  const int wave  = tid >> 5;
  const int lane  = tid & 31;
  const int ncol  = lane & 15;            // N (and A-row M) within tile
  const int khalf = lane >> 4;            // 0/1: K sub-pair for A, M+8 for C/D
  const int m0    = blockIdx.x << 4;      // batch row base

  const float* __restrict__ wg0 = ws + WG_OFF + (size_t)(0 * NTILES_G + wave) * (KK_G * 64);
  const float* __restrict__ wg1 = ws + WG_OFF + (size_t)(1 * NTILES_G + wave) * (KK_G * 64);
  const float* __restrict__ wg2 = ws + WG_OFF + (size_t)(2 * NTILES_G + wave) * (KK_G * 64);
  const float* __restrict__ wg3 = ws + WG_OFF + (size_t)(3 * NTILES_G + wave) * (KK_G * 64);
  const bool hasOut = (wave < NTILES_O);
  const float* __restrict__ wo = ws + WO_OFF + (size_t)(hasOut ? wave : 0) * (KK_H * 64);

  float bg0 = ws[BG_OFF + 0 * 288 + wave * 16 + ncol];
  float bg1 = ws[BG_OFF + 1 * 288 + wave * 16 + ncol];
  float bg2 = ws[BG_OFF + 2 * 288 + wave * 16 + ncol];
  float bg3 = ws[BG_OFF + 3 * 288 + wave * 16 + ncol];
  const float bo = ws[BO_OFF + (hasOut ? wave : 0) * 16 + ncol];

  // h_0 = 0 (incl. zero pad for K=275 column)
  for (int i = tid; i < 16 * HS; i += NTHREADS) h_lds[i] = 0.0f;
  // stage x_0 (plain path; once)
  for (int i = tid; i < 16 * INPUT; i += NTHREADS) {
    int mr = i / INPUT, k = i % INPUT;
    x_lds[0][i] = x[((size_t)(m0 + mr) * TT + 0) * INPUT + k];
  }
  v8f cst = {};
  __syncthreads();

  const int a_h = ncol * HS + 2 * khalf;      // A-frag base in h_lds (+4*kk)
  const int a_x = ncol * INPUT + 2 * khalf;   // A-frag base in x_lds (+4*kk)
  const int bfo = 2 * lane;                   // lane's slot in 64-float B frag

  for (int t = 0; t <= TT; ++t) {
    const bool doGate = (t < TT);             // grid-uniform
    const bool doOut  = hasOut && (t > 0);    // wave-uniform

    v8f acc0, acc1, acc2, acc3, oacc;
#pragma unroll
    for (int v = 0; v < 8; ++v) {
      acc0[v] = bg0; acc1[v] = bg1; acc2[v] = bg2; acc3[v] = bg3; oacc[v] = bo;
    }

    // Prefetch x_{t+1} into the other LDS buffer (drained before barrier2)
    if (t + 1 < TT) {
      const int nxt = (t + 1) & 1;
#ifdef HAVE_ASYNC_LDS
      for (int i = tid; i < (16 * INPUT) / 4; i += NTHREADS) {   // 400 x vec4
        const int mr = i / (INPUT / 4), k4 = (i % (INPUT / 4)) * 4;
        async_copy_b128(x + ((size_t)(m0 + mr) * TT + (t + 1)) * INPUT + k4,
                        &x_lds[nxt][mr * INPUT + k4]);
      }
#else
      for (int i = tid; i < 16 * INPUT; i += NTHREADS) {
        int mr = i / INPUT, k = i % INPUT;
        x_lds[nxt][i] = x[((size_t)(m0 + mr) * TT + (t + 1)) * INPUT + k];
      }
#endif
    }

    // ---- recurrent K (h_{t-1}); branch-free per-variant hot loops ----
    if (doGate) {
      if (doOut) {
#pragma unroll 3
        for (int kk = 0; kk < KK_H; ++kk) {
          v2f a = *(const v2f*)(h_lds + a_h + 4 * kk);
          acc0 = wmma4(a, *(const v2f*)(wg0 + kk * 64 + bfo), acc0);
          acc1 = wmma4(a, *(const v2f*)(wg1 + kk * 64 + bfo), acc1);
          acc2 = wmma4(a, *(const v2f*)(wg2 + kk * 64 + bfo), acc2);
          acc3 = wmma4(a, *(const v2f*)(wg3 + kk * 64 + bfo), acc3);
          oacc = wmma4(a, *(const v2f*)(wo  + kk * 64 + bfo), oacc);
        }
      } else {
#pragma unroll 3
        for (int kk = 0; kk < KK_H; ++kk) {
          v2f a = *(const v2f*)(h_lds + a_h + 4 * kk);
          acc0 = wmma4(a, *(const v2f*)(wg0 + kk * 64 + bfo), acc0);
          acc1 = wmma4(a, *(const v2f*)(wg1 + kk * 64 + bfo), acc1);
          acc2 = wmma4(a, *(const v2f*)(wg2 + kk * 64 + bfo), acc2);
          acc3 = wmma4(a, *(const v2f*)(wg3 + kk * 64 + bfo), acc3);
        }
      }
      // ---- input K (x_t) ----
      const float* __restrict__ xb = &x_lds[t & 1][0];
#pragma unroll 5
      for (int kk = 0; kk < KK_X; ++kk) {
        v2f a = *(const v2f*)(xb + a_x + 4 * kk);
        acc0 = wmma4(a, *(const v2f*)(wg0 + (KK_H + kk) * 64 + bfo), acc0);
        acc1 = wmma4(a, *(const v2f*)(wg1 + (KK_H + kk) * 64 + bfo), acc1);
        acc2 = wmma4(a, *(const v2f*)(wg2 + (KK_H + kk) * 64 + bfo), acc2);
        acc3 = wmma4(a, *(const v2f*)(wg3 + (KK_H + kk) * 64 + bfo), acc3);
      }
    } else if (doOut) {      // epilogue t == TT: out-projection only
#pragma unroll 3
      for (int kk = 0; kk < KK_H; ++kk) {
        v2f a = *(const v2f*)(h_lds + a_h + 4 * kk);
        oacc = wmma4(a, *(const v2f*)(wo + kk * 64 + bfo), oacc);
      }
    }

    // ---- store out[t-1] (D layout: VGPR v -> batch row v + 8*khalf) ----
    if (doOut) {
      const int no = wave * 16 + ncol;
      if (no < OUTD) {
#pragma unroll
        for (int v = 0; v < 8; ++v) {
          const int bi = m0 + v + 8 * khalf;
          out[((size_t)bi * TT + (t - 1)) * OUTD + no] = oacc[v];
        }
      }
    }

    // ---- gate nonlinearities, state update, publish h_t ----
    if (doGate) {
      v8f hval;
#pragma unroll
      for (int v = 0; v < 8; ++v) {
        float iv = sigmoid_f(acc0[v]);
        float fv = sigmoid_f(acc1[v]);
        float gv = tanh_f(acc2[v]);
        float ov = sigmoid_f(acc3[v]);
        float cv = fv * cst[v] + iv * gv;
        cst[v] = cv;
        hval[v] = ov * tanh_f(cv);
      }
      const int col = wave * 16 + ncol;
      const bool valid = (col < HID);
      __syncthreads();  // everyone done reading h_{t-1}
#pragma unroll
      for (int v = 0; v < 8; ++v) {
        h_lds[(v + 8 * khalf) * HS + col] = valid ? hval[v] : 0.0f;
      }
#ifdef HAVE_ASYNC_LDS
      wait_async0();    // drain async x_{t+1} -> LDS before signaling
#endif
      __syncthreads();  // h_t (and prefetched x_{t+1}) visible
    }
  }
}

extern "C" void kernel_launch(void* const* d_in, const int* in_sizes, int n_in,
                              void* d_out, int out_size, void* d_ws, size_t ws_size,
                              hipStream_t stream) {
  const float* x     = (const float*)d_in[0];
  const float* W_ih  = (const float*)d_in[1];
  const float* W_hh  = (const float*)d_in[2];
  const float* b_ih  = (const float*)d_in[3];
  const float* b_hh  = (const float*)d_in[4];
  const float* W_out = (const float*)d_in[5];
  const float* b_out = (const float*)d_in[6];
  float* ws  = (float*)d_ws;   // needs ~1.82 MB
  float* out = (float*)d_out;

  pack_kernel<<<(WS_TOTAL + 255) / 256, 256, 0, stream>>>(W_ih, W_hh, b_ih, b_hh,
                                                          W_out, b_out, ws);
  lstm_kernel<<<BATCH / 16, NTHREADS, 0, stream>>>(x, ws, out);
}